// MeshUpSample_29137058136340
// MI455X (gfx1250) — compile-verified
//
#include <hip/hip_runtime.h>
#include <hip/hip_bf16.h>

// ---------------- problem constants (match reference) ----------------
constexpr int kC  = 128;     // channels
constexpr int kNE = 50000;   // elements
constexpr int kBS = 4;       // batch
constexpr int kNN = 20000;   // nodes
constexpr float kEps   = 1e-5f;
constexpr float kSlope = 0.2f;

constexpr int kTilesTotal = (kNE + 15) / 16;                 // 3125 tiles of 16 elements
constexpr int kTPW        = 8;                               // tiles per wave
constexpr int kGroups     = (kTilesTotal + kTPW - 1) / kTPW; // 391
constexpr int kWavesTotal = kGroups * kBS * kC;              // 200192
constexpr int kWavesPerBlock = 8;                            // 256 threads
constexpr int kBlocks     = (kWavesTotal + kWavesPerBlock - 1) / kWavesPerBlock;

typedef float v2f __attribute__((ext_vector_type(2)));
typedef float v8f __attribute__((ext_vector_type(8)));

__device__ __forceinline__ float lrelu(float v) {
    return v > 0.0f ? v : kSlope * v;
}

// ---------------- zero the output (d_out is poisoned) ----------------
__global__ void zero_kernel(float* __restrict__ out, int n4) {
    int i = blockIdx.x * blockDim.x + threadIdx.x;
    if (i < n4) {
        reinterpret_cast<float4*>(out)[i] = make_float4(0.f, 0.f, 0.f, 0.f);
    }
}

// ---------------- main fused kernel ----------------
// One wave = 16 elements x 1 channel x 1 batch, looped over kTPW tiles.
//
// Transposed WMMA formulation with duplicated A rows so NO cross-lane ops are needed:
//  stage 2: A rows {0,1,8,9} = W2^T features {0,1,2,3}  ->
//           d1[0] = {feat0 | feat2}, d1[1] = {feat1 | feat3} across lane halves,
//           which IS the stage-3 B layout (K=j lower half, K=j+2 upper half).
//  stage 3: A rows {0..3} and {8..11} = W3^T corners {0..3} twice ->
//           every lane holds all 4 corners in d2[0..3]; LayerNorm is pure per-lane VALU
//           and the scatter values are local selects.
__global__ __launch_bounds__(256)
void mesh_mlp_scatter_kernel(const float* __restrict__ x,
                             const int*   __restrict__ conn,   // (NE,4) int32
                             const float* __restrict__ W1,     // (C,4)
                             const float* __restrict__ b1,     // (C,4)
                             const float* __restrict__ W2,     // (C,4,4)
                             const float* __restrict__ b2,     // (C,4)
                             const float* __restrict__ W3,     // (C,4,4)
                             const float* __restrict__ ln_w,   // (C,4)
                             const float* __restrict__ ln_b,   // (C,4)
                             float*       __restrict__ out)    // (BS,NN,C)
{
    const int lane = threadIdx.x & 31;
    const int wid  = blockIdx.x * kWavesPerBlock + (threadIdx.x >> 5);
    if (wid >= kWavesTotal) return;

    // element-group is the SLOW dimension so concurrent waves share x cachelines in L2
    const int c = wid % kC;
    const int b = (wid / kC) % kBS;
    const int g = wid / (kC * kBS);

    const int n = lane & 15;   // element index (B/N role) or A row (M role)
    const int h = lane >> 4;   // lane half

    // ---- stage-1 weights: lane needs W1/b1 at d = 2h, 2h+1 (B-operand build) ----
    const float w1_0 = W1[c * 4 + 2 * h];
    const float w1_1 = W1[c * 4 + 2 * h + 1];
    const float b1_0 = b1[c * 4 + 2 * h];
    const float b1_1 = b1[c * 4 + 2 * h + 1];

    // ---- stage-2 A operand: rows {0,1,8,9} -> W2^T features {0,1,2,3}, else 0 ----
    // A layout: lane -> M row, VGPR j -> K = j + 2h
    v2f aW2 = {0.f, 0.f};
    {
        const bool valid2 = ((n & 7) < 2);          // n in {0,1,8,9}
        const int  f      = (n & 1) + ((n >> 3) << 1);  // 0,1 -> 0,1 ; 8,9 -> 2,3
        if (valid2) {
            aW2.x = W2[c * 16 + (2 * h + 0) * 4 + f];
            aW2.y = W2[c * 16 + (2 * h + 1) * 4 + f];
        }
    }

    // ---- stage-3 A operand: rows {0..3} AND {8..11} -> W3^T corners {0..3}, else 0 ----
    v2f aW3 = {0.f, 0.f};
    {
        const bool valid3 = ((n & 4) == 0);         // n in {0..3, 8..11}
        const int  f      = n & 3;
        if (valid3) {
            aW3.x = W3[c * 16 + (2 * h + 0) * 4 + f];
            aW3.y = W3[c * 16 + (2 * h + 1) * 4 + f];
        }
    }

    // ---- stage-2 C operand: biases follow the duplicated-row mapping ----
    // VGPR 0: row0 (lower) = b2[0], row8 (upper) = b2[2]
    // VGPR 1: row1 (lower) = b2[1], row9 (upper) = b2[3]
    v8f cbias;
#pragma unroll
    for (int r = 0; r < 8; ++r) cbias[r] = 0.f;
    cbias[0] = (h == 0) ? b2[c * 4 + 0] : b2[c * 4 + 2];
    cbias[1] = (h == 0) ? b2[c * 4 + 1] : b2[c * 4 + 3];

    v8f czero;
#pragma unroll
    for (int r = 0; r < 8; ++r) czero[r] = 0.f;

    // ---- LayerNorm affine params: wave-uniform (corner = result index) ----
    const float lnw0 = ln_w[c * 4 + 0], lnw1 = ln_w[c * 4 + 1];
    const float lnw2 = ln_w[c * 4 + 2], lnw3 = ln_w[c * 4 + 3];
    const float lnb0 = ln_b[c * 4 + 0], lnb1 = ln_b[c * 4 + 1];
    const float lnb2 = ln_b[c * 4 + 2], lnb3 = ln_b[c * 4 + 3];

    const int tbase = g * kTPW;
#pragma unroll
    for (int i = 0; i < kTPW; ++i) {
        const int t = tbase + i;
        if (t >= kTilesTotal) break;

        const int e = t * 16 + n;                 // this lane's element
        float s = 0.f;
        if (e < kNE) s = x[(size_t)(b * kNE + e) * kC + c];

        // ---- stage 1: h1^T directly in B layout: VGPR j, half h -> K = j + 2h ----
        v2f bh1;
        bh1.x = lrelu(fmaf(s, w1_0, b1_0));
        bh1.y = lrelu(fmaf(s, w1_1, b1_1));

        // ---- stage 2: h2^T = W2^T(dup rows) x h1^T + b2 ----
        v8f d1 = __builtin_amdgcn_wmma_f32_16x16x4_f32(
            false, aW2, false, bh1, (short)0, cbias, false, false);

        // d1[0] = {feat0 | feat2}, d1[1] = {feat1 | feat3} -> lrelu gives stage-3 B directly
        v2f bh2;
        bh2.x = lrelu(d1[0]);
        bh2.y = lrelu(d1[1]);

        // ---- stage 3: h3^T = W3^T(dup rows) x h2^T ----
        v8f d2 = __builtin_amdgcn_wmma_f32_16x16x4_f32(
            false, aW3, false, bh2, (short)0, czero, false, false);

        // ---- LayerNorm: every lane holds all 4 corners in d2[0..3] ----
        const float v0 = d2[0], v1 = d2[1], v2 = d2[2], v3 = d2[3];
        const float ss = (v0 + v1) + (v2 + v3);
        const float qq = fmaf(v0, v0, fmaf(v1, v1, fmaf(v2, v2, v3 * v3)));
        const float mu  = ss * 0.25f;
        const float var = fmaf(-mu, mu, qq * 0.25f);
        const float inv = rsqrtf(var + kEps);
        const float hn0 = fmaf((v0 - mu) * inv, lnw0, lnb0);
        const float hn1 = fmaf((v1 - mu) * inv, lnw1, lnb1);
        const float hn2 = fmaf((v2 - mu) * inv, lnw2, lnb2);
        const float hn3 = fmaf((v3 - mu) * inv, lnw3, lnb3);

        // ---- scatter: lane handles corners {2h, 2h+1} of its element, local selects ----
        const float val0 = (h == 0) ? hn0 : hn2;
        const float val1 = (h == 0) ? hn1 : hn3;

        if (e < kNE) {
            // coalesced: lane reads conn[e*4 + 2h .. e*4 + 2h+1] as one int2
            const int2 nodes = reinterpret_cast<const int2*>(conn)[(size_t)e * 2 + h];
            float* outb = out + (size_t)b * kNN * kC + c;
            __hip_atomic_fetch_add(outb + (size_t)nodes.x * kC, val0,
                                   __ATOMIC_RELAXED, __HIP_MEMORY_SCOPE_AGENT);
            __hip_atomic_fetch_add(outb + (size_t)nodes.y * kC, val1,
                                   __ATOMIC_RELAXED, __HIP_MEMORY_SCOPE_AGENT);
        }
    }
}

extern "C" void kernel_launch(void* const* d_in, const int* in_sizes, int n_in,
                              void* d_out, int out_size, void* d_ws, size_t ws_size,
                              hipStream_t stream) {
    const float* x    = (const float*)d_in[0];
    const int*   conn = (const int*)  d_in[1];   // JAX default x64-disabled -> int32
    const float* W1   = (const float*)d_in[2];
    const float* b1   = (const float*)d_in[3];
    const float* W2   = (const float*)d_in[4];
    const float* b2   = (const float*)d_in[5];
    const float* W3   = (const float*)d_in[6];
    const float* lnw  = (const float*)d_in[7];
    const float* lnb  = (const float*)d_in[8];
    float* out = (float*)d_out;

    // zero the (poisoned) output, BS*NN*C floats, vectorized float4
    const int n4 = (kBS * kNN * kC) / 4;
    zero_kernel<<<(n4 + 255) / 256, 256, 0, stream>>>(out, n4);

    mesh_mlp_scatter_kernel<<<kBlocks, 256, 0, stream>>>(
        x, conn, W1, b1, W2, b2, W3, lnw, lnb, out);
}